// OrbitalAEVComputer_1211180777778
// MI455X (gfx1250) — compile-verified
//
#include <hip/hip_runtime.h>
#include <math.h>

// ---------------------------------------------------------------------------
// OrbitalAEVComputer for MI455X (gfx1250, wave32).
// Store-bandwidth-bound: 448 MB of output vs 4.4 MB input -> optimize for
// coalesced non-temporal b128 stores; use one V_WMMA_F32_16X16X4_F32 per atom
// to compute the 12x12 Gram matrix (all 66 pair dot products at once).
// ---------------------------------------------------------------------------

typedef __attribute__((ext_vector_type(2))) float v2f;
typedef __attribute__((ext_vector_type(4))) float v4f;
typedef __attribute__((ext_vector_type(8))) float v8f;

#define NPAIR 66

// species==8 -> row 1 (O), else row 0 (H)
__constant__ float c_mus[2][9] = {
    {-0.0013891633279463555f, 0.052817133273316774f, 0.139812833597163f,
      0.04197082575153015f, 0.0f, 0.0f, 0.0f, 0.0f, 0.0f},
    { 0.3869724094534165f, 1.1195726605859562f, 4.597656411500856f,
      4.18116756869605f, -0.777753667038269f, 0.8991094564592526f,
      0.25287442792706855f, 0.0f, 0.0f}};
__constant__ float c_sgs[2][9] = {
    { 0.014984132924379046f, 0.06597323056314336f, 0.05229082868259066f,
      0.10317217531939492f, 1.0f, 1.0f, 1.0f, 1.0f, 1.0f},
    { 0.01648580936203453f, 0.04360467004470585f, 0.10939958699162289f,
      0.1452881891452822f, 0.24744978852384295f, 0.6757096584771374f,
      0.15629951845443388f, 1.0f, 1.0f}};

__global__ __launch_bounds__(128) void
aev_kernel(const float* __restrict__ coef, const int* __restrict__ species,
           float* __restrict__ out, int natoms_total) {
  const int atom = blockIdx.x;
  if (atom >= natoms_total) return;
  const int t = threadIdx.x;
  const float* __restrict__ c = coef + (size_t)atom * 45;
  float* __restrict__ o = out + (size_t)atom * 4560;

  __shared__ float sn[9];        // normalized s coefficients
  __shared__ float om[12][3];    // orbital moment vectors
  __shared__ float dist[12];
  __shared__ float rinv[12];     // 0 if zero-masked
  __shared__ float pc[NPAIR];    // 0.9999 * cos(angle)
  __shared__ float pav[NPAIR];   // average distance
  __shared__ float f1t[NPAIR][8];
  __shared__ float f2t[NPAIR][8];

  // -------- phase 1: per-atom scalars into LDS --------
  if (t < 9) {
    const int row = (species[atom] == 8) ? 1 : 0;
    sn[t] = (c[t] - c_mus[row][t]) / c_sgs[row][t];
  }
  if (t < 12) {
    float x, y, z;
    if (t < 4) {                      // p block: coef[9:21] as 4x3
      x = c[9 + t * 3 + 0]; y = c[9 + t * 3 + 1]; z = c[9 + t * 3 + 2];
    } else if (t < 8) {               // d reordered cols (0,2,5)
      const int a = t - 4;
      x = c[21 + a * 6 + 0]; y = c[21 + a * 6 + 2]; z = c[21 + a * 6 + 5];
    } else {                          // d reordered cols (4,3,1)
      const int a = t - 8;
      x = c[21 + a * 6 + 4]; y = c[21 + a * 6 + 3]; z = c[21 + a * 6 + 1];
    }
    om[t][0] = x; om[t][1] = y; om[t][2] = z;
    const float dd = sqrtf(x * x + y * y + z * z);
    dist[t] = dd;
    const bool zero =
        (fabsf(x) < 1e-12f) && (fabsf(y) < 1e-12f) && (fabsf(z) < 1e-12f);
    rinv[t] = zero ? 0.0f : (1.0f / dd);
  }
  __syncthreads();

  // -------- phase 2a: wave 0 computes the 12x12 Gram matrix via WMMA --------
  if (t < 32) {                       // wave-uniform branch: EXEC all ones
    const int lane = t;
    const int m16 = lane & 15;
    v2f a;
    if (m16 < 12) {
      if (lane < 16) { a.x = om[m16][0]; a.y = om[m16][1]; }  // K=0,1
      else           { a.x = om[m16][2]; a.y = 0.0f;       }  // K=2,3(pad)
    } else {
      a.x = 0.0f; a.y = 0.0f;         // rows 12..15 zero-padded
    }
    v8f acc = {};
    // D = A(16x4) x B(4x16) + 0 ; B operand mirrors A layout, and since
    // B = A^T for a Gram product the per-lane register contents coincide.
    acc = __builtin_amdgcn_wmma_f32_16x16x4_f32(
        false, a, false, a, (short)0, acc, false, false);
#pragma unroll
    for (int r = 0; r < 8; ++r) {
      const int m = r + ((lane >= 16) ? 8 : 0);   // D row
      const int n = lane & 15;                    // D col
      if (n < 12 && m < n) {
        const int p = m * 11 - (m * (m - 1)) / 2 + n - m - 1;  // triu index
        pc[p]  = acc[r] * rinv[m] * rinv[n] * 0.9999f;
        pav[p] = 0.5f * (dist[m] + dist[n]);
      }
    }
  }

  // -------- phase 2b (all threads, overlaps wave0's WMMA): s/r features ----
  for (int idx = t; idx < 144; idx += 128) {
    const int i = idx >> 4, j = idx & 15;
    const float sh = -4.0f + (float)j * (8.0f / 15.0f);
    const float d1 = sn[i] - sh;
    __builtin_nontemporal_store(__expf(-4.0f * d1 * d1), &o[idx]);
  }
  for (int idx = t; idx < 192; idx += 128) {
    const int i = idx >> 4, j = idx & 15;
    const float sh = (float)j * (2.0f / 15.0f);
    const float d1 = dist[i] - sh;
    __builtin_nontemporal_store(__expf(-4.0f * d1 * d1), &o[144 + idx]);
  }
  __syncthreads();

  // -------- phase 3: f1/f2 tables (66 pairs x 8 each) --------
  // angle = arccos(c'); cos(angle - Z) = c'*cos(Z) + sqrt(1-c'^2)*sin(Z)
  const float PI7 = 0.44879895051282760549f;  // pi/7
  for (int idx = t; idx < NPAIR * 16; idx += 128) {
    const int p = idx >> 4, q = idx & 15;
    if (q < 8) {
      const float cc = pc[p];
      const float ss = sqrtf(fmaxf(0.0f, 1.0f - cc * cc));
      float sz, cz;
      __sincosf((float)q * PI7, &sz, &cz);
      const float cosd = cc * cz + ss * sz;
      const float y = 0.5f * (1.0f + cosd);
      const float y2 = y * y, y4 = y2 * y2;
      f1t[p][q] = y4 * y4;              // ((1+cos)/2)^8 by squaring
    } else {
      const int a = q - 8;
      const float sh = (float)a * (2.0f / 7.0f);
      const float d1 = pav[p] - sh;
      f2t[p][a] = __expf(-8.0f * d1 * d1);
    }
  }
  __syncthreads();

  // -------- phase 4: angular outer products, 1056 coalesced b128 stores ----
  v4f* __restrict__ o4 = (v4f*)(o + 336);
  for (int k = t; k < NPAIR * 16; k += 128) {
    const int p = k >> 4;
    const int r = k & 15;
    const int z = r >> 1;
    const int half = r & 1;
    const float f = 2.0f * f1t[p][z];
    const float* f2p = &f2t[p][half * 4];
    v4f v;
    v.x = f * f2p[0]; v.y = f * f2p[1]; v.z = f * f2p[2]; v.w = f * f2p[3];
    __builtin_nontemporal_store(v, &o4[k]);
  }
}

extern "C" void kernel_launch(void* const* d_in, const int* in_sizes, int n_in,
                              void* d_out, int out_size, void* d_ws,
                              size_t ws_size, hipStream_t stream) {
  const float* coef = (const float*)d_in[0];
  const int* species = (const int*)d_in[1];
  float* out = (float*)d_out;
  const int natoms_total = in_sizes[1];  // nconf * natoms (species count)
  dim3 grid((unsigned)natoms_total), block(128);
  hipLaunchKernelGGL(aev_kernel, grid, block, 0, stream, coef, species, out,
                     natoms_total);
}